// GCN2Layer_1056561955280
// MI455X (gfx1250) — compile-verified
//
#include <hip/hip_runtime.h>
#include <hip/hip_bf16.h>

#define N_NODES 50000
#define N_EDGES 800000
#define HDIM    128
#define NCLS    10
#define NGRAPH  512
#define BN_EPS  1e-5f
#define NB_SCAN ((N_NODES + 255) / 256)   // 196 scan blocks

typedef float v2f __attribute__((ext_vector_type(2)));
typedef float v8f __attribute__((ext_vector_type(8)));

// ---- in-degree histogram (int counts in `cnt`) ----
__global__ void deg_hist_kernel(const int* __restrict__ dst, int* __restrict__ cnt, int E) {
    int e = blockIdx.x * blockDim.x + threadIdx.x;
    if (e < E) atomicAdd(&cnt[dst[e]], 1);
}

// ---- dis[i] = (deg+1)^{-1/2} from int counts ----
__global__ void deg_to_dis_kernel(const int* __restrict__ cnt, float* __restrict__ dis, int n) {
    int i = blockIdx.x * blockDim.x + threadIdx.x;
    if (i < n) dis[i] = rsqrtf((float)cnt[i] + 1.0f);
}

// ---- scan stage A: per-256-block exclusive scan (LDS ladder), block totals out ----
__global__ __launch_bounds__(256) void scan_block_kernel(const int* __restrict__ cnt,
                                                         int* __restrict__ off,
                                                         int* __restrict__ bsum, int n) {
    __shared__ int sh[256];
    int t = threadIdx.x;
    int i = blockIdx.x * 256 + t;
    int v = (i < n) ? cnt[i] : 0;
    sh[t] = v;
    __syncthreads();
#pragma unroll
    for (int d = 1; d < 256; d <<= 1) {
        int x = (t >= d) ? sh[t - d] : 0;
        __syncthreads();
        sh[t] += x;
        __syncthreads();
    }
    if (i < n) off[i] = sh[t] - v;            // exclusive within block
    if (t == 0) bsum[blockIdx.x] = sh[255];   // block total
}

// ---- scan stage B: exclusive scan of block sums (single block) ----
__global__ __launch_bounds__(256) void scan_top_kernel(int* __restrict__ bsum, int nb) {
    __shared__ int sh[256];
    int t = threadIdx.x;
    int v = (t < nb) ? bsum[t] : 0;
    sh[t] = v;
    __syncthreads();
#pragma unroll
    for (int d = 1; d < 256; d <<= 1) {
        int x = (t >= d) ? sh[t - d] : 0;
        __syncthreads();
        sh[t] += x;
        __syncthreads();
    }
    if (t < nb) bsum[t] = sh[t] - v;          // exclusive
}

// ---- scan stage C: add block bases; init cursors; close CSR ----
__global__ void scan_fix_kernel(int* __restrict__ off, const int* __restrict__ bsum,
                                int* __restrict__ cursor, int n, int E) {
    int i = blockIdx.x * blockDim.x + threadIdx.x;
    if (i < n) {
        int o = off[i] + bsum[i >> 8];
        off[i] = o;
        cursor[i] = o;
    }
    if (i == 0) off[n] = E;
}

// ---- counting-sort placement: sorted_src grouped by dst ----
__global__ void edge_place_kernel(const int* __restrict__ src, const int* __restrict__ dst,
                                  int* __restrict__ cursor, int* __restrict__ ssrc, int E) {
    int e = blockIdx.x * blockDim.x + threadIdx.x;
    if (e < E) {
        int p = atomicAdd(&cursor[dst[e]], 1);
        ssrc[p] = src[e];
    }
}

// ---- C[Mx128] = A[Mx128] @ B[128x128] via V_WMMA_F32_16X16X4_F32 ----
// One wave per 16x16 tile; block = 4 waves x 2 tiles = full 128-col strip.
__global__ __launch_bounds__(128) void gemm128_wmma(const float* __restrict__ A,
                                                    const float* __restrict__ B,
                                                    float* __restrict__ C) {
    const int K = 128, NO = 128;
    int wave = threadIdx.x >> 5;
    int lane = threadIdx.x & 31;
    int row0 = blockIdx.x << 4;
    int m  = lane & 15;
    int kh = (lane >> 4) << 1;                 // 0 or 2
    const float* arow = A + (size_t)(row0 + m) * K;
#pragma unroll
    for (int t = 0; t < 2; ++t) {
        int n0 = ((wave << 1) + t) << 4;
        v8f acc = {};
#pragma unroll
        for (int k0 = 0; k0 < K; k0 += 4) {
            v2f av, bv;
            av[0] = arow[k0 + kh];
            av[1] = arow[k0 + kh + 1];
            bv[0] = B[(k0 + kh)     * NO + n0 + m];
            bv[1] = B[(k0 + kh + 1) * NO + n0 + m];
            acc = __builtin_amdgcn_wmma_f32_16x16x4_f32(
                false, av, false, bv, (short)0, acc, false, false);
        }
        int mbase = row0 + ((lane >> 4) << 3);
#pragma unroll
        for (int r = 0; r < 8; ++r)
            C[(size_t)(mbase + r) * NO + n0 + m] = acc[r];
    }
}

// ---- fused per-node gather-aggregate + self-loop + bias + BN + ReLU ----
// One wave per node: 32 lanes x float4 = 128 channels held in registers.
__global__ __launch_bounds__(256) void aggregate_bn_relu_kernel(
        const float* __restrict__ h, const int* __restrict__ off, const int* __restrict__ ssrc,
        const float* __restrict__ dis, const float* __restrict__ bias,
        const float* __restrict__ gamma, const float* __restrict__ beta,
        const float* __restrict__ mean, const float* __restrict__ var,
        float* __restrict__ act, int n) {
    int node = blockIdx.x * 8 + (threadIdx.x >> 5);
    if (node >= n) return;
    int lane = threadIdx.x & 31;

    float di = dis[node];
    // self-loop term: h[node] * dis^2
    float4 acc = ((const float4*)(h + (size_t)node * HDIM))[lane];
    float dii = di * di;
    acc.x *= dii; acc.y *= dii; acc.z *= dii; acc.w *= dii;

    int e0 = off[node], e1 = off[node + 1];
    for (int e = e0; e < e1; ++e) {
        int s = ssrc[e];
        float w = dis[s] * di;
        float4 v = ((const float4*)(h + (size_t)s * HDIM))[lane];
        acc.x += v.x * w; acc.y += v.y * w; acc.z += v.z * w; acc.w += v.w * w;
    }

    float4 b4  = ((const float4*)bias )[lane];
    float4 g4  = ((const float4*)gamma)[lane];
    float4 be4 = ((const float4*)beta )[lane];
    float4 m4  = ((const float4*)mean )[lane];
    float4 v4  = ((const float4*)var  )[lane];
    float4 o;
    o.x = fmaxf(g4.x * (acc.x + b4.x - m4.x) * rsqrtf(v4.x + BN_EPS) + be4.x, 0.0f);
    o.y = fmaxf(g4.y * (acc.y + b4.y - m4.y) * rsqrtf(v4.y + BN_EPS) + be4.y, 0.0f);
    o.z = fmaxf(g4.z * (acc.z + b4.z - m4.z) * rsqrtf(v4.z + BN_EPS) + be4.z, 0.0f);
    o.w = fmaxf(g4.w * (acc.w + b4.w - m4.w) * rsqrtf(v4.w + BN_EPS) + be4.w, 0.0f);
    ((float4*)(act + (size_t)node * HDIM))[lane] = o;
}

// ---- global mean pool: sums via atomics (only 6.4M, negligible) ----
__global__ void pool_sum_kernel(const float* __restrict__ act, const int* __restrict__ batch,
                                float* __restrict__ gsum, int n) {
    int tid = blockIdx.x * blockDim.x + threadIdx.x;
    int q = tid & 31;
    int i = tid >> 5;
    if (i >= n) return;
    int b = batch[i];
    float4 v = ((const float4*)(act + (size_t)i * HDIM))[q];
    float* gp = gsum + (size_t)b * HDIM + (q << 2);
    atomicAdd(gp + 0, v.x);
    atomicAdd(gp + 1, v.y);
    atomicAdd(gp + 2, v.z);
    atomicAdd(gp + 3, v.w);
}

__global__ void pool_cnt_kernel(const int* __restrict__ batch, float* __restrict__ gcnt, int n) {
    int i = blockIdx.x * blockDim.x + threadIdx.x;
    if (i < n) atomicAdd(&gcnt[batch[i]], 1.0f);
}

// ---- out[g,c] = (gsum[g,:] @ Wlin[:,c]) / max(cnt,1) + blin[c] ----
__global__ void final_linear_kernel(const float* __restrict__ gsum, const float* __restrict__ gcnt,
                                    const float* __restrict__ Wlin, const float* __restrict__ blin,
                                    float* __restrict__ out) {
    int tid = blockIdx.x * blockDim.x + threadIdx.x;
    if (tid >= NGRAPH * NCLS) return;
    int g = tid / NCLS, c = tid % NCLS;
    const float* gp = gsum + (size_t)g * HDIM;
    float acc = 0.0f;
#pragma unroll 8
    for (int k = 0; k < HDIM; ++k) acc += gp[k] * Wlin[k * NCLS + c];
    out[tid] = acc / fmaxf(gcnt[g], 1.0f) + blin[c];
}

extern "C" void kernel_launch(void* const* d_in, const int* in_sizes, int n_in,
                              void* d_out, int out_size, void* d_ws, size_t ws_size,
                              hipStream_t stream) {
    const float* x     = (const float*)d_in[0];
    const int*   edge  = (const int*)d_in[1];
    const int*   src   = edge;
    const int*   dst   = edge + N_EDGES;
    const int*   batch = (const int*)d_in[2];
    const float* W1   = (const float*)d_in[3];
    const float* b1   = (const float*)d_in[4];
    const float* g1   = (const float*)d_in[5];
    const float* be1  = (const float*)d_in[6];
    const float* m1   = (const float*)d_in[7];
    const float* v1   = (const float*)d_in[8];
    const float* W2   = (const float*)d_in[9];
    const float* b2   = (const float*)d_in[10];
    const float* g2   = (const float*)d_in[11];
    const float* be2  = (const float*)d_in[12];
    const float* m2   = (const float*)d_in[13];
    const float* v2   = (const float*)d_in[14];
    const float* Wlin = (const float*)d_in[15];
    const float* blin = (const float*)d_in[16];
    float* out = (float*)d_out;

    // workspace: hbuf[N*H] | act[N*H] | dis[N] | gsum[G*H] | gcnt[G]
    //          | off[N+1] | cursor[N] | bsum[256] | ssrc[E]
    float* hbuf   = (float*)d_ws;
    float* act    = hbuf + (size_t)N_NODES * HDIM;
    float* dis    = act  + (size_t)N_NODES * HDIM;
    float* gsum   = dis  + N_NODES;
    float* gcnt   = gsum + (size_t)NGRAPH * HDIM;
    int*   off    = (int*)(gcnt + NGRAPH);
    int*   cursor = off + (N_NODES + 1);
    int*   bsum   = cursor + N_NODES;
    int*   ssrc   = bsum + 256;

    // ---- degree histogram -> dis, CSR offsets (scan), counting sort by dst ----
    hipMemsetAsync(cursor, 0, N_NODES * sizeof(int), stream);   // cursor doubles as cnt
    deg_hist_kernel<<<(N_EDGES + 255) / 256, 256, 0, stream>>>(dst, cursor, N_EDGES);
    deg_to_dis_kernel<<<(N_NODES + 255) / 256, 256, 0, stream>>>(cursor, dis, N_NODES);
    scan_block_kernel<<<NB_SCAN, 256, 0, stream>>>(cursor, off, bsum, N_NODES);
    scan_top_kernel<<<1, 256, 0, stream>>>(bsum, NB_SCAN);
    scan_fix_kernel<<<(N_NODES + 255) / 256, 256, 0, stream>>>(off, bsum, cursor, N_NODES, N_EDGES);
    edge_place_kernel<<<(N_EDGES + 255) / 256, 256, 0, stream>>>(src, dst, cursor, ssrc, N_EDGES);

    // ---- layer 1: GEMM (WMMA f32) then fused gather-aggregate+BN+ReLU ----
    gemm128_wmma<<<N_NODES / 16, 128, 0, stream>>>(x, W1, hbuf);
    aggregate_bn_relu_kernel<<<(N_NODES + 7) / 8, 256, 0, stream>>>(
        hbuf, off, ssrc, dis, b1, g1, be1, m1, v1, act, N_NODES);

    // ---- layer 2 ----
    gemm128_wmma<<<N_NODES / 16, 128, 0, stream>>>(act, W2, hbuf);
    aggregate_bn_relu_kernel<<<(N_NODES + 7) / 8, 256, 0, stream>>>(
        hbuf, off, ssrc, dis, b2, g2, be2, m2, v2, act, N_NODES);

    // ---- global mean pool + classifier ----
    hipMemsetAsync(gsum, 0, (size_t)(NGRAPH * HDIM + NGRAPH) * sizeof(float), stream);
    pool_sum_kernel<<<(N_NODES * 32 + 255) / 256, 256, 0, stream>>>(act, batch, gsum, N_NODES);
    pool_cnt_kernel<<<(N_NODES + 255) / 256, 256, 0, stream>>>(batch, gcnt, N_NODES);
    final_linear_kernel<<<(NGRAPH * NCLS + 255) / 256, 256, 0, stream>>>(gsum, gcnt, Wlin, blin, out);
}